// MultiLayerGATv2_68547678045007
// MI455X (gfx1250) — compile-verified
//
#include <hip/hip_runtime.h>
#include <math.h>

#define HID    96
#define INDIM  128
#define NLAYER 2

typedef float v2f __attribute__((ext_vector_type(2)));
typedef float v8f __attribute__((ext_vector_type(8)));

// ---- float <-> order-preserving uint (for atomicMax on floats incl. negatives) ----
__device__ __forceinline__ unsigned ordenc(float f) {
    unsigned u = __float_as_uint(f);
    return (u & 0x80000000u) ? ~u : (u | 0x80000000u);
}
__device__ __forceinline__ float orddec(unsigned m) {
    unsigned u = (m & 0x80000000u) ? (m ^ 0x80000000u) : ~m;
    return __uint_as_float(u);
}

// =====================================================================
// C[M x 96] = A[M x K] @ B[K x 96]   (fp32 WMMA 16x16x4, wave32)
// block = 192 threads = 6 waves; wave w owns cols [16w, 16w+16) of a
// 16-row strip. K is a compile-time constant so the K-loop fully
// unrolls and every load is base-pointer + immediate offset.
// Rows are CLAMPED (not predicated) so the loop has no exec masking;
// only the final stores are guarded.
// =====================================================================
template<int K>
__global__ __launch_bounds__(192)
void k_gemm_n96(const float* __restrict__ A, const float* __restrict__ B,
                float* __restrict__ C, int M)
{
    const int lane = threadIdx.x & 31;
    const int wave = threadIdx.x >> 5;          // 0..5
    const int row0 = blockIdx.x * 16;
    const int kh   = (lane >> 4) * 2;           // 0 or 2 (K-pair selector)
    int row = row0 + (lane & 15);
    if (row >= M) row = M - 1;                  // clamp: loads always in-bounds
    const int col  = wave * 16 + (lane & 15);

    // per-lane base pointers; all loop accesses become immediate offsets
    const float2* __restrict__ a2   = (const float2*)(A + (size_t)row * K) + (kh >> 1);
    const float*  __restrict__ bcol = B + (size_t)kh * HID + col;

    v8f acc = {};
    #pragma unroll
    for (int k = 0; k < K; k += 4) {
        const float2 t = a2[k >> 1];            // A[row][k+kh], A[row][k+kh+1]
        v2f a; a.x = t.x; a.y = t.y;
        v2f b;
        b.x = bcol[(size_t)k * HID];            // B[k+kh  ][col]
        b.y = bcol[(size_t)k * HID + HID];      // B[k+kh+1][col]
        acc = __builtin_amdgcn_wmma_f32_16x16x4_f32(
                  false, a, false, b, (short)0, acc, false, false);
    }

    const int mbase = 8 * (lane >> 4);
    #pragma unroll
    for (int r = 0; r < 8; ++r) {
        const int rr = row0 + mbase + r;
        if (rr < M) C[(size_t)rr * HID + col] = acc[r];
    }
}

// =====================================================================
// Wave-per-row LayerNorm + activation (+ optional bias in, residual out)
// act==0 : ReLU ; act==1 : ELU
// =====================================================================
__global__ __launch_bounds__(256)
void k_postln(const float* __restrict__ in, const float* __restrict__ bias,
              const float* __restrict__ g, const float* __restrict__ b,
              const float* __restrict__ resid, float* __restrict__ out,
              int N, int act)
{
    const int lane = threadIdx.x & 31;
    const int row  = blockIdx.x * 8 + (threadIdx.x >> 5);
    if (row >= N) return;
    const float* p = in + (size_t)row * HID;
    float v0 = p[lane], v1 = p[lane + 32], v2 = p[lane + 64];
    if (bias) { v0 += bias[lane]; v1 += bias[lane + 32]; v2 += bias[lane + 64]; }
    float s = v0 + v1 + v2;
    for (int o = 16; o; o >>= 1) s += __shfl_xor(s, o, 32);
    const float mu = s * (1.f / HID);
    const float d0 = v0 - mu, d1 = v1 - mu, d2 = v2 - mu;
    float q = d0 * d0 + d1 * d1 + d2 * d2;
    for (int o = 16; o; o >>= 1) q += __shfl_xor(q, o, 32);
    const float rs = 1.f / sqrtf(q * (1.f / HID) + 1e-5f);
    float y0 = d0 * rs * g[lane]      + b[lane];
    float y1 = d1 * rs * g[lane + 32] + b[lane + 32];
    float y2 = d2 * rs * g[lane + 64] + b[lane + 64];
    if (act == 0) {
        y0 = fmaxf(y0, 0.f); y1 = fmaxf(y1, 0.f); y2 = fmaxf(y2, 0.f);
    } else {
        y0 = (y0 > 0.f) ? y0 : (expf(y0) - 1.f);
        y1 = (y1 > 0.f) ? y1 : (expf(y1) - 1.f);
        y2 = (y2 > 0.f) ? y2 : (expf(y2) - 1.f);
    }
    float r0 = 0.f, r1 = 0.f, r2 = 0.f;
    if (resid) {
        const float* rp = resid + (size_t)row * HID;
        r0 = rp[lane]; r1 = rp[lane + 32]; r2 = rp[lane + 64];
    }
    float* op = out + (size_t)row * HID;
    op[lane] = r0 + y0; op[lane + 32] = r1 + y1; op[lane + 64] = r2 + y2;
}

// =====================================================================
// Per-call scratch init: segment max = -inf (encoded), denom = 0, accum = 0
// =====================================================================
__global__ void k_init(unsigned* __restrict__ maxb, float* __restrict__ denom,
                       float* __restrict__ accum, int N)
{
    const long long i = (long long)blockIdx.x * blockDim.x + threadIdx.x;
    if (i < (long long)N * HID) accum[i] = 0.f;
    if (i < 2LL * N) { maxb[i] = 0x007FFFFFu; denom[i] = 0.f; }  // enc(-inf)
}

// =====================================================================
// Pass 1: wave per edge — score[e][h] = <leakyrelu(xl[s]+xr[d]), att_h>,
// atomicMax segment max per (dst, head). Channels 0-47 = head0, 48-95 = head1.
// =====================================================================
__global__ __launch_bounds__(256)
void k_edge_score(const int* __restrict__ srcA, const int* __restrict__ dstA,
                  int E, int Et, const float* __restrict__ xl,
                  const float* __restrict__ xr, const float* __restrict__ att,
                  float* __restrict__ score, unsigned* __restrict__ maxb)
{
    const int lane = threadIdx.x & 31;
    const int e = blockIdx.x * 8 + (threadIdx.x >> 5);
    if (e >= Et) return;
    int s, d;
    if (e < E) { s = srcA[e]; d = dstA[e]; } else { s = d = e - E; }
    const float* pl = xl + (size_t)s * HID;
    const float* pr = xr + (size_t)d * HID;
    float v0 = pl[lane]      + pr[lane];
    float v1 = pl[lane + 32] + pr[lane + 32];
    float v2 = pl[lane + 64] + pr[lane + 64];
    v0 = ((v0 > 0.f) ? v0 : 0.2f * v0) * att[lane];
    v1 = ((v1 > 0.f) ? v1 : 0.2f * v1) * att[lane + 32];
    v2 = ((v2 > 0.f) ? v2 : 0.2f * v2) * att[lane + 64];
    float p0 = v0 + ((lane < 16) ? v1 : 0.f);     // head 0 (c in [0,48))
    float p1 = v2 + ((lane < 16) ? 0.f : v1);     // head 1 (c in [48,96))
    for (int o = 16; o; o >>= 1) {
        p0 += __shfl_xor(p0, o, 32);
        p1 += __shfl_xor(p1, o, 32);
    }
    if (lane == 0) {
        score[2 * (size_t)e]     = p0;
        score[2 * (size_t)e + 1] = p1;
        atomicMax(&maxb[2 * d],     ordenc(p0));
        atomicMax(&maxb[2 * d + 1], ordenc(p1));
    }
}

// =====================================================================
// Pass 2: thread per (edge, head) — p = exp(score - max[dst]), sum into denom
// =====================================================================
__global__ void k_edge_exp(const int* __restrict__ dstA, int E, int Et,
                           float* __restrict__ score,
                           const unsigned* __restrict__ maxb,
                           float* __restrict__ denom)
{
    const long long idx = (long long)blockIdx.x * blockDim.x + threadIdx.x;
    if (idx >= 2LL * Et) return;
    const int e = (int)(idx >> 1), hh = (int)(idx & 1);
    const int d = (e < E) ? dstA[e] : (e - E);
    const float m = orddec(maxb[2 * d + hh]);
    const float pv = expf(score[idx] - m);
    score[idx] = pv;
    atomicAdd(&denom[2 * d + hh], pv);
}

// =====================================================================
// Pass 3: wave per edge — accum[dst] += alpha_h * xl[src]
// =====================================================================
__global__ __launch_bounds__(256)
void k_edge_scatter(const int* __restrict__ srcA, const int* __restrict__ dstA,
                    int E, int Et, const float* __restrict__ xl,
                    const float* __restrict__ prob,
                    const float* __restrict__ denom, float* __restrict__ accum)
{
    const int lane = threadIdx.x & 31;
    const int e = blockIdx.x * 8 + (threadIdx.x >> 5);
    if (e >= Et) return;
    int s, d;
    if (e < E) { s = srcA[e]; d = dstA[e]; } else { s = d = e - E; }
    const float a0 = prob[2 * (size_t)e]     / denom[2 * d];
    const float a1 = prob[2 * (size_t)e + 1] / denom[2 * d + 1];
    const float* pl = xl + (size_t)s * HID;
    float* pa = accum + (size_t)d * HID;
    atomicAdd(pa + lane,      a0 * pl[lane]);
    atomicAdd(pa + lane + 32, ((lane < 16) ? a0 : a1) * pl[lane + 32]);
    atomicAdd(pa + lane + 64, a1 * pl[lane + 64]);
}

// =====================================================================
extern "C" void kernel_launch(void* const* d_in, const int* in_sizes, int n_in,
                              void* d_out, int out_size, void* d_ws, size_t ws_size,
                              hipStream_t stream)
{
    (void)n_in; (void)out_size; (void)ws_size;
    const float* x   = (const float*)d_in[0];
    const int*   ei  = (const int*)  d_in[1];
    const float* Wp  = (const float*)d_in[2];
    const float* bp  = (const float*)d_in[3];
    const float* g0  = (const float*)d_in[4];
    const float* b0  = (const float*)d_in[5];
    const float* Wl  = (const float*)d_in[6];
    const float* Wr  = (const float*)d_in[7];
    const float* att = (const float*)d_in[8];
    const float* gs  = (const float*)d_in[9];
    const float* bs  = (const float*)d_in[10];

    const int N  = in_sizes[0] / INDIM;
    const int E  = in_sizes[1] / 2;
    const int Et = E + N;
    float* out = (float*)d_out;

    // scratch carve (all fp32, ~85 MB for N=50k, Et=850k)
    float* ws = (float*)d_ws;
    float* h     = ws;              ws += (size_t)N * HID;
    float* xl    = ws;              ws += (size_t)N * HID;
    float* xr    = ws;              ws += (size_t)N * HID;
    float* accum = ws;              ws += (size_t)N * HID;
    float* score = ws;              ws += (size_t)Et * 2;
    unsigned* maxb = (unsigned*)ws; ws += (size_t)N * 2;
    float* denom = ws;

    const dim3 gB(192), gG((N + 15) / 16);
    const dim3 lnB(256), lnG((N + 7) / 8);
    const dim3 egB(256), egG((Et + 7) / 8);
    const dim3 exG((unsigned)((2LL * Et + 255) / 256));
    const dim3 inG((unsigned)(((long long)N * HID + 255) / 256));

    // input proj -> LN -> ReLU
    k_gemm_n96<INDIM><<<gG, gB, 0, stream>>>(x, Wp, accum, N);
    k_postln<<<lnG, lnB, 0, stream>>>(accum, bp, g0, b0, nullptr, h, N, /*relu*/0);

    for (int l = 0; l < NLAYER; ++l) {
        k_gemm_n96<HID><<<gG, gB, 0, stream>>>(h, Wl + (size_t)l * HID * HID, xl, N);
        k_gemm_n96<HID><<<gG, gB, 0, stream>>>(h, Wr + (size_t)l * HID * HID, xr, N);
        k_init<<<inG, 256, 0, stream>>>(maxb, denom, accum, N);
        k_edge_score<<<egG, egB, 0, stream>>>(ei, ei + E, E, Et, xl, xr,
                                              att + (size_t)l * HID, score, maxb);
        k_edge_exp<<<exG, 256, 0, stream>>>(ei + E, E, Et, score, maxb, denom);
        k_edge_scatter<<<egG, egB, 0, stream>>>(ei, ei + E, E, Et, xl, score,
                                                denom, accum);
        k_postln<<<lnG, lnB, 0, stream>>>(accum, nullptr,
                                          gs + (size_t)l * HID, bs + (size_t)l * HID,
                                          h, (l == NLAYER - 1) ? out : h, N, /*elu*/1);
    }
}